// CXTRNN_18880676233539
// MI455X (gfx1250) — compile-verified
//
#include <hip/hip_runtime.h>

// Problem constants (match reference)
#define SEQ   512
#define BATCH 4096
#define DIMS  3
#define DIMY  3
#define DIMZ  6
#define RANK  6
#define HID   50
#define KG    13      // K-groups of 4 covering K = 0..51 (HID padded to 52)
#define XSTR  68      // LDS row stride (floats) for state buffer: even, avoids bank conflicts
#define ALPHA 0.5f

typedef float v2f __attribute__((ext_vector_type(2)));
typedef float v8f __attribute__((ext_vector_type(8)));

__device__ __forceinline__ v8f wmma_f32_k4(v2f a, v2f b, v8f c) {
    // D(16x16,f32) = A(16x4,f32) * B(4x16,f32) + C
    return __builtin_amdgcn_wmma_f32_16x16x4_f32(false, a, false, b, (short)0, c, false, false);
}

// tanh: native v_tanh_f32 if the builtin exists on this toolchain, else exp+rcp
__device__ __forceinline__ float fast_tanh(float x) {
#if __has_builtin(__builtin_amdgcn_tanhf)
    return __builtin_amdgcn_tanhf(x);
#else
    x = fminf(15.0f, fmaxf(-15.0f, x));
    float e = __expf(2.0f * x);
    float r = __builtin_amdgcn_rcpf(e + 1.0f);
    return fmaf(-2.0f, r, 1.0f);
#endif
}

// sigmoid via v_exp_f32 + v_rcp_f32
__device__ __forceinline__ float fast_sigmoid(float x) {
    return __builtin_amdgcn_rcpf(1.0f + __expf(-x));
}

__global__ __launch_bounds__(32) void cxtrnn_wmma_kernel(
    const float* __restrict__ s,   const float* __restrict__ z,
    const float* __restrict__ U,   const float* __restrict__ V,
    const float* __restrict__ Wi,  const float* __restrict__ bi,
    const float* __restrict__ Wo,  const float* __restrict__ bo,
    const float* __restrict__ nmW, const float* __restrict__ nmb,
    float* __restrict__ out)
{
    // Per-wave private LDS (single wave per workgroup; no barriers needed,
    // intra-wave ordering via s_wait_dscnt).
    __shared__ float ldsX[16 * XSTR];  // state X [16 rows][padded 64 cols]
    __shared__ float ldsP[16 * 16];    // P = tanh(X)@V  [16 rows][16 cols]

    const int lane = (int)threadIdx.x;
    const int col  = lane & 15;        // N-column in B/C/D layouts; row in A layout
    const int hi   = lane >> 4;        // half-wave selector
    const bool hib = (hi != 0);
    const int b0   = (int)blockIdx.x * 16;   // first batch row of this wave
    const int brow = b0 + col;               // batch row this lane serves (A-layout)

    // ---------------- constant weights in WMMA B-layout ----------------
    // B-layout (4x16 f32): vgpr j, half hi holds B[K = 4g + 2*hi + j][N = col]
    v2f Bv[KG];                 // V: [52,16], B[k][n] = V[k*RANK+n], pads = 0
#pragma unroll
    for (int g = 0; g < KG; ++g) {
        int k0 = 4 * g + 2 * hi;
        float a0 = (k0     < HID && col < RANK) ? V[k0 * RANK + col]       : 0.0f;
        float a1 = (k0 + 1 < HID && col < RANK) ? V[(k0 + 1) * RANK + col] : 0.0f;
        v2f t; t[0] = a0; t[1] = a1; Bv[g] = t;
    }
    // ALPHA folded into Bu and Bw so the WMMA chain directly yields
    // acc = ALPHA*(tmp + inp); state update is then a single FMA per element.
    v2f Bu[4][2];               // ALPHA * U^T: [8,64], B[k][h] = a*U[h*RANK+k], pads = 0
#pragma unroll
    for (int t4 = 0; t4 < 4; ++t4) {
        int h = 16 * t4 + col;
#pragma unroll
        for (int g = 0; g < 2; ++g) {
            int k0 = 4 * g + 2 * hi;
            float a0 = (h < HID && k0     < RANK) ? ALPHA * U[h * RANK + k0]     : 0.0f;
            float a1 = (h < HID && k0 + 1 < RANK) ? ALPHA * U[h * RANK + k0 + 1] : 0.0f;
            v2f t; t[0] = a0; t[1] = a1; Bu[t4][g] = t;
        }
    }
    v2f Bw[4];                  // ALPHA * [W_in^T ; b_in]: [4,64], row3 = bias
#pragma unroll
    for (int t4 = 0; t4 < 4; ++t4) {
        int h  = 16 * t4 + col;
        int k0 = 2 * hi;        // K = {0,1} or {2,3}
        float a0 = 0.0f, a1 = 0.0f;
        if (h < HID) {
            a0 = ALPHA * Wi[h * DIMS + k0];                                  // k0 in {0,2}
            a1 = ALPHA * ((k0 + 1 < DIMS) ? Wi[h * DIMS + k0 + 1] : bi[h]);  // K=1 or K=3(bias)
        }
        v2f t; t[0] = a0; t[1] = a1; Bw[t4] = t;
    }
    v2f Bo[KG];                 // [W_out^T ; b_out]: [52,16], row50 = b_out
#pragma unroll
    for (int g = 0; g < KG; ++g) {
        int k0 = 4 * g + 2 * hi;
        float a0 = 0.0f, a1 = 0.0f;
        if (col < DIMY) {
            a0 = (k0     < HID) ? Wo[col * HID + k0]     : ((k0     == HID) ? bo[col] : 0.0f);
            a1 = (k0 + 1 < HID) ? Wo[col * HID + k0 + 1] : ((k0 + 1 == HID) ? bo[col] : 0.0f);
        }
        v2f t; t[0] = a0; t[1] = a1; Bo[g] = t;
    }
    // gating params: uniform -> scalar loads / SGPR residency (confirmed in asm)
    float nmw[RANK][DIMZ], nmbv[RANK];
#pragma unroll
    for (int r = 0; r < RANK; ++r) {
        nmbv[r] = nmb[r];
#pragma unroll
        for (int i = 0; i < DIMZ; ++i) nmw[r][i] = nmW[r * DIMZ + i];
    }

    // ---------------- recurrent state ----------------
    v8f XD[4];                  // X in D-layout, 4 column tiles of 16
#pragma unroll
    for (int t4 = 0; t4 < 4; ++t4) { v8f zz = {0,0,0,0,0,0,0,0}; XD[t4] = zz; }
    float Atan[2 * KG];         // tanh(X) in A-layout (vgpr pairs per K-group)
#pragma unroll
    for (int i = 0; i < 2 * KG; ++i) Atan[i] = 0.0f;

    const v8f vzero = {0,0,0,0,0,0,0,0};

#pragma unroll 1
    for (int t = 0; t < SEQ; ++t) {
        // -------- stream z_t, s_t for this lane's batch row --------
        const float* zr = z + ((size_t)t * BATCH + (size_t)brow) * DIMZ;
        const float* sr = s + ((size_t)t * BATCH + (size_t)brow) * DIMS;
        float zv[DIMZ];
#pragma unroll
        for (int i = 0; i < DIMZ; ++i) zv[i] = zr[i];
        float sv0 = sr[0], sv1 = sr[1], sv2 = sr[2];
        float zsum = 0.0f;
#pragma unroll
        for (int i = 0; i < DIMZ; ++i) zsum += zv[i];
        // gating: named scalars (no dynamic array indexing -> no per-iter compares)
        float g0, g1, g2, g3, g4, g5;
        {
            float a0 = nmbv[0], a1 = nmbv[1], a2 = nmbv[2];
            float a3 = nmbv[3], a4 = nmbv[4], a5 = nmbv[5];
#pragma unroll
            for (int i = 0; i < DIMZ; ++i) {
                a0 = fmaf(nmw[0][i], zv[i], a0);
                a1 = fmaf(nmw[1][i], zv[i], a1);
                a2 = fmaf(nmw[2][i], zv[i], a2);
                a3 = fmaf(nmw[3][i], zv[i], a3);
                a4 = fmaf(nmw[4][i], zv[i], a4);
                a5 = fmaf(nmw[5][i], zv[i], a5);
            }
            g0 = fast_sigmoid(a0); g1 = fast_sigmoid(a1); g2 = fast_sigmoid(a2);
            g3 = fast_sigmoid(a3); g4 = fast_sigmoid(a4); g5 = fast_sigmoid(a5);
        }
        // per-lane selected gains for A-layout K slots (single cndmask each)
        const float ga0 = hib ? g2 : g0;     // K = 2*hi
        const float ga1 = hib ? g3 : g1;     // K = 2*hi + 1
        const float ga2 = hib ? 0.0f : g4;   // K = 4 + 2*hi  (6 -> pad 0)
        const float ga3 = hib ? 0.0f : g5;   // K = 5 + 2*hi  (7 -> pad 0)

        // -------- P = tanh(X) @ V : 13 chained f32 WMMAs --------
        v8f P = vzero;
#pragma unroll
        for (int g = 0; g < KG; ++g) {
            v2f a; a[0] = Atan[2 * g]; a[1] = Atan[2 * g + 1];
            P = wmma_f32_k4(a, Bv[g], P);
        }

        // -------- D-layout -> A-layout for P via LDS, scale by g --------
        // (cols 6..15 of P are exactly 0; store all 16 cols: no EXEC divergence)
#pragma unroll
        for (int v = 0; v < 8; ++v) ldsP[(v + 8 * hi) * 16 + col] = P[v];
        asm volatile("s_wait_dscnt 0" ::: "memory");
        v2f Ap0, Ap1;
        {
            float p0 = ldsP[col * 16 + 2 * hi];
            float p1 = ldsP[col * 16 + 2 * hi + 1];
            float p2 = ldsP[col * 16 + 4 + 2 * hi];
            float p3 = ldsP[col * 16 + 4 + 2 * hi + 1];
            Ap0[0] = p0 * ga0; Ap0[1] = p1 * ga1;
            Ap1[0] = p2 * ga2; Ap1[1] = p3 * ga3;   // hi half -> *0 (pad)
        }
        // A_s = [zsum*s_t | zsum] (bias folded into K=3; ALPHA folded into Bw)
        v2f As;
        As[0] = zsum * (hib ? sv2 : sv0);          // K=0 -> s0 ; K=2 -> s2
        As[1] = hib ? zsum : (zsum * sv1);         // K=1 -> s1 ; K=3 -> 1 (bias row)

        // -------- acc = a*((P*g)@U^T + inp) ; X = fma(X, 1-a, acc) --------
#pragma unroll
        for (int t4 = 0; t4 < 4; ++t4) {
            v8f acc = wmma_f32_k4(Ap0, Bu[t4][0], vzero);
            acc = wmma_f32_k4(Ap1, Bu[t4][1], acc);
            acc = wmma_f32_k4(As,  Bw[t4],    acc);
            v8f x = XD[t4];
#pragma unroll
            for (int v = 0; v < 8; ++v) x[v] = fmaf(x[v], 1.0f - ALPHA, acc[v]);
            XD[t4] = x;
#pragma unroll
            for (int v = 0; v < 8; ++v)
                ldsX[(v + 8 * hi) * XSTR + 16 * t4 + col] = x[v];
        }
        asm volatile("s_wait_dscnt 0" ::: "memory");

        // -------- rebuild A-layout tanh(x_new) (reused next step) --------
#pragma unroll
        for (int g = 0; g < KG; ++g) {
            int k0 = 4 * g + 2 * hi;
            Atan[2 * g]     = fast_tanh(ldsX[col * XSTR + k0]);
            Atan[2 * g + 1] = fast_tanh(ldsX[col * XSTR + k0 + 1]);
        }

        // -------- out = [zsum*tanh(x_new) | zsum] @ [Wo^T ; bo] --------
        v8f OD = vzero;
#pragma unroll
        for (int g = 0; g < KG; ++g) {
            float a0 = zsum * Atan[2 * g];
            float a1 = zsum * Atan[2 * g + 1];
            if (g == KG - 1) {            // K=50 -> bias row weight zsum, K=51 -> 0
                a0 = hib ? zsum : a0;
                a1 = hib ? 0.0f : a1;
            }
            v2f a; a[0] = a0; a[1] = a1;
            OD = wmma_f32_k4(a, Bo[g], OD);
        }

        // -------- store out[t, b0+8*hi+v, col] for col<3 --------
        if (col < DIMY) {
            size_t base = (size_t)t * BATCH * DIMY + (size_t)(b0 + 8 * hi) * DIMY + col;
#pragma unroll
            for (int v = 0; v < 8; ++v) out[base + (size_t)v * DIMY] = OD[v];
        }
    }
}

extern "C" void kernel_launch(void* const* d_in, const int* in_sizes, int n_in,
                              void* d_out, int out_size, void* d_ws, size_t ws_size,
                              hipStream_t stream) {
    (void)in_sizes; (void)n_in; (void)out_size; (void)d_ws; (void)ws_size;
    const float* s   = (const float*)d_in[0];
    const float* z   = (const float*)d_in[1];
    const float* U   = (const float*)d_in[2];
    const float* V   = (const float*)d_in[3];
    const float* Wi  = (const float*)d_in[4];
    const float* bi  = (const float*)d_in[5];
    const float* Wo  = (const float*)d_in[6];
    const float* bo  = (const float*)d_in[7];
    const float* nmW = (const float*)d_in[8];
    const float* nmb = (const float*)d_in[9];
    float* out = (float*)d_out;

    dim3 grid(BATCH / 16);   // 256 independent waves, one per workgroup -> spread over WGPs
    dim3 block(32);          // one wave32
    hipLaunchKernelGGL(cxtrnn_wmma_kernel, grid, block, 0, stream,
                       s, z, U, V, Wi, bi, Wo, bo, nmW, nmb, out);
}